// EnergyXL_55877524521083
// MI455X (gfx1250) — compile-verified
//
#include <hip/hip_runtime.h>
#include <math.h>

// ---------------------------------------------------------------------------
// Problem constants (match reference: NMOL=64, MOLSIZE=100, n=400, RANK=4)
// ---------------------------------------------------------------------------
#define NMOL   64
#define MS     100
#define NDIM   400
#define NN2    ((size_t)NDIM * NDIM)          // 160000
#define MATSZ  ((size_t)NMOL * NN2)           // 10,240,000 floats per full batch
#define BLKSZ  ((size_t)NMOL * MS * MS * 16)  // 10,240,000 floats (4x4 blocks)
#define NATOM  (NMOL * MS)                    // 6400
#define RANK   4
#define NSWEEP 10

#define TEMPC  1500.0f
#define KBC    8.61739e-05f
#define BETA   (1.0f / (KBC * TEMPC))

typedef float v2f __attribute__((ext_vector_type(2)));
typedef float v8f __attribute__((ext_vector_type(8)));

__device__ __constant__ int   c_TRI_I[10] = {0, 0, 1, 0, 1, 2, 0, 1, 2, 3};
__device__ __constant__ int   c_TRI_J[10] = {0, 1, 1, 2, 2, 2, 3, 3, 3, 3};
__device__ __constant__ float c_WGT[10]   = {1.f, 2.f, 1.f, 2.f, 2.f, 1.f, 2.f, 2.f, 2.f, 1.f};
__device__ __constant__ int   c_IND[4][4] = {{0, 1, 3, 6}, {1, 2, 4, 7}, {3, 4, 5, 8}, {6, 7, 8, 9}};

// ---------------------------------------------------------------------------
// Elementwise helpers
// ---------------------------------------------------------------------------
__global__ void k_fill(float* __restrict__ p, float v, size_t n) {
  size_t i = (size_t)blockIdx.x * blockDim.x + threadIdx.x;
  if (i < n) p[i] = v;
}

// out = a - b
__global__ void k_sub(float* __restrict__ out, const float* __restrict__ a,
                      const float* __restrict__ b, size_t n) {
  size_t i = (size_t)blockIdx.x * blockDim.x + threadIdx.x;
  if (i < n) out[i] = a[i] - b[i];
}

// v -= dots[b] * vj   (b = i / NN2)
__global__ void k_proj(float* __restrict__ v, const float* __restrict__ vj,
                       const float* __restrict__ dots) {
  size_t i = (size_t)blockIdx.x * blockDim.x + threadIdx.x;
  if (i < MATSZ) {
    int b = (int)(i / NN2);
    v[i] -= dots[b] * vj[i];
  }
}

// v *= 1/sqrt(norm2[b])
__global__ void k_scalenorm(float* __restrict__ v, const float* __restrict__ n2) {
  size_t i = (size_t)blockIdx.x * blockDim.x + threadIdx.x;
  if (i < MATSZ) {
    int b = (int)(i / NN2);
    v[i] *= 1.0f / sqrtf(n2[b]);
  }
}

// T[b,i,j] = Q[b,i,j] * f[b,j]   (column scale -> Q * diag(f))
__global__ void k_colscale(float* __restrict__ T, const float* __restrict__ Q,
                           const float* __restrict__ f) {
  size_t i = (size_t)blockIdx.x * blockDim.x + threadIdx.x;
  if (i < MATSZ) {
    int b = (int)(i / NN2);
    int j = (int)(i % NDIM);
    T[i] = Q[i] * f[b * NDIM + j];
  }
}

// ---------------------------------------------------------------------------
// Fock build
// ---------------------------------------------------------------------------
// Pb[((b*MS+I)*MS+J)*16 + a*4+c] = src[b,4I+a,4J+c]
__global__ void k_blockify(float* __restrict__ Pb, const float* __restrict__ src) {
  size_t idx = (size_t)blockIdx.x * blockDim.x + threadIdx.x;
  if (idx >= BLKSZ) return;
  int ac = (int)(idx & 15);
  size_t bij = idx >> 4;
  int a = ac >> 2, c = ac & 3;
  int J = (int)(bij % MS);
  size_t bI = bij / MS;
  int I = (int)(bI % MS);
  int b = (int)(bI / MS);
  Pb[idx] = src[(size_t)b * NN2 + (size_t)(4 * I + a) * NDIM + 4 * J + c];
}

// Diagonal one-center terms: one thread per atom
__global__ void k_diag_fock(float* __restrict__ Fb, const float* __restrict__ Pb,
                            const float* __restrict__ gss, const float* __restrict__ gpp,
                            const float* __restrict__ gsp, const float* __restrict__ gp2,
                            const float* __restrict__ hsp, const int* __restrict__ maskd) {
  int t = blockIdx.x * blockDim.x + threadIdx.x;
  if (t >= NATOM) return;
  int bd = maskd[t];
  const float* Pd = Pb + (size_t)bd * 16;
  float* Fd = Fb + (size_t)bd * 16;
  float P00 = Pd[0];
  float Pp[4] = {0.f, Pd[5], Pd[10], Pd[15]};
  float Pptot = Pp[1] + Pp[2] + Pp[3];
  float vgss = gss[t], vgpp = gpp[t], vgsp = gsp[t], vgp2 = gp2[t], vhsp = hsp[t];
  Fd[0] += 0.5f * P00 * vgss + Pptot * (vgsp - 0.5f * vhsp);
  for (int i = 1; i < 4; ++i) {
    Fd[i * 5] += P00 * (vgsp - 0.5f * vhsp) + 0.5f * Pp[i] * vgpp +
                 (Pptot - Pp[i]) * (1.25f * vgp2 - 0.25f * vgpp);
    Fd[i] += Pd[i] * (1.5f * vhsp - 0.5f * vgsp);
  }
  Fd[1 * 4 + 2] += Pd[1 * 4 + 2] * (0.75f * vgpp - 1.25f * vgp2);
  Fd[1 * 4 + 3] += Pd[1 * 4 + 3] * (0.75f * vgpp - 1.25f * vgp2);
  Fd[2 * 4 + 3] += Pd[2 * 4 + 3] * (0.75f * vgpp - 1.25f * vgp2);
}

// Two-center Coulomb: one thread per pair, atomic adds into diagonal blocks
__global__ void k_pair_coulomb(float* __restrict__ Fb, const float* __restrict__ Pb,
                               const float* __restrict__ w, const int* __restrict__ maskd,
                               const int* __restrict__ idxi, const int* __restrict__ idxj,
                               int npair) {
  int p = blockIdx.x * blockDim.x + threadIdx.x;
  if (p >= npair) return;
  int bi = maskd[idxi[p]];
  int bj = maskd[idxj[p]];
  float PA[10], PB[10], suma[10], sumb[10];
#pragma unroll
  for (int t = 0; t < 10; ++t) {
    int off = c_TRI_I[t] * 4 + c_TRI_J[t];
    PA[t] = Pb[(size_t)bi * 16 + off] * c_WGT[t];
    PB[t] = Pb[(size_t)bj * 16 + off] * c_WGT[t];
    suma[t] = 0.f;
  }
  const float* wp = w + (size_t)p * 100;
#pragma unroll
  for (int i = 0; i < 10; ++i) {
    float pai = PA[i], sb = 0.f;
#pragma unroll
    for (int j = 0; j < 10; ++j) {
      float wv = wp[i * 10 + j];
      suma[j] += pai * wv;
      sb += PB[j] * wv;
    }
    sumb[i] = sb;
  }
#pragma unroll
  for (int t = 0; t < 10; ++t) {
    int off = c_TRI_I[t] * 4 + c_TRI_J[t];
    atomicAdd(&Fb[(size_t)bj * 16 + off], suma[t]);
    atomicAdd(&Fb[(size_t)bi * 16 + off], sumb[t]);
  }
}

// Exchange on off-diagonal pair blocks: one thread per pair (blocks unique, no atomics)
__global__ void k_exchange(float* __restrict__ Fb, const float* __restrict__ Pb,
                           const float* __restrict__ w, const int* __restrict__ mask,
                           int npair) {
  int p = blockIdx.x * blockDim.x + threadIdx.x;
  if (p >= npair) return;
  int bp = mask[p];
  const float* Pm = Pb + (size_t)bp * 16;
  const float* wp = w + (size_t)p * 100;
  float acc[16];
#pragma unroll
  for (int t = 0; t < 16; ++t) acc[t] = 0.f;
#pragma unroll
  for (int a = 0; a < 4; ++a)
#pragma unroll
    for (int c = 0; c < 4; ++c) {
      float pm = -0.5f * Pm[a * 4 + c];
#pragma unroll
      for (int i = 0; i < 4; ++i) {
        int wi = c_IND[i][a] * 10;
#pragma unroll
        for (int j = 0; j < 4; ++j) acc[i * 4 + j] += pm * wp[wi + c_IND[j][c]];
      }
    }
  float* Fd = Fb + (size_t)bp * 16;
#pragma unroll
  for (int t = 0; t < 16; ++t) Fd[t] += acc[t];
}

// Unblock + symmetrize from upper triangle
__global__ void k_unblock_sym(float* __restrict__ Fout, const float* __restrict__ Fb) {
  size_t idx = (size_t)blockIdx.x * blockDim.x + threadIdx.x;
  if (idx >= MATSZ) return;
  int b = (int)(idx / NN2);
  size_t rem = idx % NN2;
  int i = (int)(rem / NDIM), j = (int)(rem % NDIM);
  int ii = (i <= j) ? i : j;
  int jj = (i <= j) ? j : i;
  int I = ii >> 2, a = ii & 3, J = jj >> 2, c = jj & 3;
  Fout[idx] = Fb[(((size_t)b * MS + I) * MS + J) * 16 + a * 4 + c];
}

// ---------------------------------------------------------------------------
// Batched Jacobi eigensolver (one workgroup per molecule, global-memory matrix)
// ---------------------------------------------------------------------------
__global__ void k_init_eig(float* __restrict__ A, float* __restrict__ Q,
                           const float* __restrict__ F) {
  size_t idx = (size_t)blockIdx.x * blockDim.x + threadIdx.x;
  if (idx >= MATSZ) return;
  size_t rem = idx % NN2;
  int i = (int)(rem / NDIM), j = (int)(rem % NDIM);
  A[idx] = F[idx];
  Q[idx] = (i == j) ? 1.0f : 0.0f;
}

__device__ inline void jac_pair(int r, int k, int& p, int& q) {
  const int m = NDIM - 1;  // 399
  if (k == 0) { p = NDIM - 1; q = r % m; }
  else { p = (r + k) % m; q = (r - k + m) % m; }
  if (p < q) { int t = p; p = q; q = t; }
}

__global__ void __launch_bounds__(1024) k_jacobi(float* __restrict__ Ag, float* __restrict__ Qg) {
  const int b = blockIdx.x;
  float* A = Ag + (size_t)b * NN2;
  float* Q = Qg + (size_t)b * NN2;
  __shared__ float cs_c[NDIM / 2];
  __shared__ float cs_s[NDIM / 2];
  const int tid = threadIdx.x, nt = blockDim.x;
  for (int sweep = 0; sweep < NSWEEP; ++sweep) {
    for (int r = 0; r < NDIM - 1; ++r) {
      // rotations for the 200 disjoint pairs of this round
      for (int k = tid; k < NDIM / 2; k += nt) {
        int p, q;
        jac_pair(r, k, p, q);
        float app = A[(size_t)p * NDIM + p];
        float aqq = A[(size_t)q * NDIM + q];
        float apq = A[(size_t)p * NDIM + q];
        float c = 1.f, s = 0.f;
        if (fabsf(apq) > 1e-13f) {
          float tau = (aqq - app) / (2.0f * apq);
          float t = copysignf(1.0f, tau) / (fabsf(tau) + sqrtf(1.0f + tau * tau));
          c = 1.0f / sqrtf(1.0f + t * t);
          s = t * c;
        }
        cs_c[k] = c; cs_s[k] = s;
      }
      __syncthreads();
      // row pass: A <- G^T A
      for (int t = tid; t < (NDIM / 2) * NDIM; t += nt) {
        int k = t / NDIM, j = t % NDIM;
        int p, q;
        jac_pair(r, k, p, q);
        float c = cs_c[k], s = cs_s[k];
        float ap = A[(size_t)p * NDIM + j], aq = A[(size_t)q * NDIM + j];
        A[(size_t)p * NDIM + j] = c * ap - s * aq;
        A[(size_t)q * NDIM + j] = s * ap + c * aq;
      }
      __syncthreads();
      // column pass: A <- A G, Q <- Q G
      for (int t = tid; t < (NDIM / 2) * NDIM; t += nt) {
        int k = t / NDIM, i = t % NDIM;
        int p, q;
        jac_pair(r, k, p, q);
        float c = cs_c[k], s = cs_s[k];
        float aip = A[(size_t)i * NDIM + p], aiq = A[(size_t)i * NDIM + q];
        A[(size_t)i * NDIM + p] = c * aip - s * aiq;
        A[(size_t)i * NDIM + q] = s * aip + c * aiq;
        float qip = Q[(size_t)i * NDIM + p], qiq = Q[(size_t)i * NDIM + q];
        Q[(size_t)i * NDIM + p] = c * qip - s * qiq;
        Q[(size_t)i * NDIM + q] = s * qip + c * qiq;
      }
      __syncthreads();
    }
  }
}

__global__ void k_diag_extract(float* __restrict__ e, const float* __restrict__ A) {
  int i = blockIdx.x * blockDim.x + threadIdx.x;
  if (i >= NMOL * NDIM) return;
  int b = i / NDIM, d = i % NDIM;
  e[i] = A[(size_t)b * NN2 + (size_t)d * NDIM + d];
}

// ---------------------------------------------------------------------------
// Fermi occupation by bisection + entropy (one workgroup per molecule)
// ---------------------------------------------------------------------------
__global__ void __launch_bounds__(512) k_fermi(const float* __restrict__ e,
                                               const int* __restrict__ nocc,
                                               float* __restrict__ f,
                                               float* __restrict__ eent,
                                               float* __restrict__ muv) {
  const int b = blockIdx.x;
  const float* eb = e + b * NDIM;
  __shared__ float red[512];
  __shared__ float s_lo, s_hi, s_mid;
  int tid = threadIdx.x;
  float ev = (tid < NDIM) ? eb[tid] : 3.0e38f;
  red[tid] = ev; __syncthreads();
  for (int s = 256; s > 0; s >>= 1) { if (tid < s) red[tid] = fminf(red[tid], red[tid + s]); __syncthreads(); }
  if (tid == 0) s_lo = red[0] - 1.0f;
  __syncthreads();
  ev = (tid < NDIM) ? eb[tid] : -3.0e38f;
  red[tid] = ev; __syncthreads();
  for (int s = 256; s > 0; s >>= 1) { if (tid < s) red[tid] = fmaxf(red[tid], red[tid + s]); __syncthreads(); }
  if (tid == 0) s_hi = red[0] + 1.0f;
  __syncthreads();
  float target = (float)nocc[b];
  for (int it = 0; it < 60; ++it) {
    if (tid == 0) s_mid = 0.5f * (s_lo + s_hi);
    __syncthreads();
    float mid = s_mid;
    float occ = (tid < NDIM) ? 1.0f / (1.0f + expf(BETA * (eb[tid] - mid))) : 0.0f;
    red[tid] = occ; __syncthreads();
    for (int s = 256; s > 0; s >>= 1) { if (tid < s) red[tid] += red[tid + s]; __syncthreads(); }
    if (tid == 0) { if (red[0] > target) s_hi = s_mid; else s_lo = s_mid; }
    __syncthreads();
  }
  if (tid == 0) { s_mid = 0.5f * (s_lo + s_hi); muv[b] = s_mid; }
  __syncthreads();
  float mu = s_mid;
  float ent = 0.0f;
  if (tid < NDIM) {
    float fv = 1.0f / (1.0f + expf(BETA * (eb[tid] - mu)));
    f[b * NDIM + tid] = fv;
    float fc = fminf(fmaxf(fv, 1e-12f), 1.0f - 1e-12f);
    ent = -(fc * logf(fc) + (1.0f - fc) * logf(1.0f - fc));
  }
  red[tid] = ent; __syncthreads();
  for (int s = 256; s > 0; s >>= 1) { if (tid < s) red[tid] += red[tid + s]; __syncthreads(); }
  if (tid == 0) eent[b] = -2.0f * TEMPC * (KBC * red[0]);
}

// ---------------------------------------------------------------------------
// Batched f32 WMMA GEMM: C[b] = alpha * opA(A[b]) * opB(B[b]); 400x400x400,
// one wave per 16x16 output tile, K stepped by 4 through V_WMMA_F32_16X16X4_F32
// ---------------------------------------------------------------------------
template <int OPA, int OPB>
__global__ void __launch_bounds__(32) k_gemm(const float* __restrict__ A,
                                             const float* __restrict__ B,
                                             float* __restrict__ C, float alpha) {
  const int tile = blockIdx.x;                 // 0..624
  const int b = blockIdx.y;
  const int row0 = (tile / (NDIM / 16)) * 16;
  const int col0 = (tile % (NDIM / 16)) * 16;
  const size_t off = (size_t)b * NN2;
  const float* Ab = A + off;
  const float* Bb = B + off;
  float* Cb = C + off;
  const int lane = threadIdx.x & 31;
  const int half = lane >> 4;   // selects K pair {0,1} vs {2,3}
  const int l16 = lane & 15;    // M index (A) / N index (B)
  v8f acc = {};
  for (int k0 = 0; k0 < NDIM; k0 += 4) {
    v2f af, bf;
    if (OPA == 0) {
      const float* pA = Ab + (size_t)(row0 + l16) * NDIM + k0 + 2 * half;
      af.x = pA[0]; af.y = pA[1];
      __builtin_prefetch(pA + 16, 0, 1);
    } else {
      af.x = Ab[(size_t)(k0 + 2 * half + 0) * NDIM + row0 + l16];
      af.y = Ab[(size_t)(k0 + 2 * half + 1) * NDIM + row0 + l16];
    }
    if (OPB == 0) {
      bf.x = Bb[(size_t)(k0 + 2 * half + 0) * NDIM + col0 + l16];
      bf.y = Bb[(size_t)(k0 + 2 * half + 1) * NDIM + col0 + l16];
    } else {
      const float* pB = Bb + (size_t)(col0 + l16) * NDIM + k0 + 2 * half;
      bf.x = pB[0]; bf.y = pB[1];
      __builtin_prefetch(pB + 16, 0, 1);
    }
    acc = __builtin_amdgcn_wmma_f32_16x16x4_f32(false, af, false, bf, (short)0, acc,
                                                false, false);
  }
#pragma unroll
  for (int r = 0; r < 8; ++r) {
    int row = row0 + r + 8 * half;
    Cb[(size_t)row * NDIM + col0 + l16] = alpha * acc[r];
  }
}

// ---------------------------------------------------------------------------
// Response helpers
// ---------------------------------------------------------------------------
// F1 *= R(e,f) elementwise
__global__ void k_rapply(float* __restrict__ F1, const float* __restrict__ e,
                         const float* __restrict__ f) {
  size_t idx = (size_t)blockIdx.x * blockDim.x + threadIdx.x;
  if (idx >= MATSZ) return;
  int b = (int)(idx / NN2);
  size_t rem = idx % NN2;
  int i = (int)(rem / NDIM), j = (int)(rem % NDIM);
  float ei = e[b * NDIM + i], ej = e[b * NDIM + j];
  float fi = f[b * NDIM + i], fj = f[b * NDIM + j];
  float de = ei - ej;
  float R;
  if (fabsf(de) < 1e-7f) {
    float fm = 0.5f * (fi + fj);
    R = -BETA * fm * (1.0f - fm);
  } else {
    R = (fi - fj) / de;
  }
  F1[idx] *= R;
}

// Gram-Schmidt projection coefficient: dots[b] = sum_ij v[b,j,i]*vj[b,i,j]
__global__ void __launch_bounds__(256) k_dot_tr(const float* __restrict__ v,
                                                const float* __restrict__ vj,
                                                float* __restrict__ dots) {
  const int b = blockIdx.x;
  const float* vb = v + (size_t)b * NN2;
  const float* wb = vj + (size_t)b * NN2;
  __shared__ float red[256];
  float acc = 0.f;
  for (int t = threadIdx.x; t < (int)NN2; t += 256) {
    int i = t / NDIM, j = t % NDIM;
    acc += vb[(size_t)j * NDIM + i] * wb[t];
  }
  red[threadIdx.x] = acc; __syncthreads();
  for (int s = 128; s > 0; s >>= 1) { if (threadIdx.x < s) red[threadIdx.x] += red[threadIdx.x + s]; __syncthreads(); }
  if (threadIdx.x == 0) dots[b] = red[0];
}

// Frobenius norm squared per batch
__global__ void __launch_bounds__(256) k_fro(const float* __restrict__ v,
                                             float* __restrict__ n2) {
  const int b = blockIdx.x;
  const float* vb = v + (size_t)b * NN2;
  __shared__ float red[256];
  float acc = 0.f;
  for (int t = threadIdx.x; t < (int)NN2; t += 256) acc += vb[t] * vb[t];
  red[threadIdx.x] = acc; __syncthreads();
  for (int s = 128; s > 0; s >>= 1) { if (threadIdx.x < s) red[threadIdx.x] += red[threadIdx.x + s]; __syncthreads(); }
  if (threadIdx.x == 0) n2[b] = red[0];
}

// O[b,I,J] = sum_ij W_I[b,j,i]*W_J[b,i,j]; tr[b,J] = sum_ij W_J[b,j,i]*dDS[b,i,j]
__global__ void __launch_bounds__(256) k_OW(const float* __restrict__ W,
                                            const float* __restrict__ dDS,
                                            float* __restrict__ O, float* __restrict__ tr) {
  const int b = blockIdx.x;
  __shared__ float red[256];
  float aO[4][4], aT[4];
#pragma unroll
  for (int I = 0; I < 4; ++I) { aT[I] = 0.f;
#pragma unroll
    for (int J = 0; J < 4; ++J) aO[I][J] = 0.f; }
  for (int t = threadIdx.x; t < (int)NN2; t += 256) {
    int i = t / NDIM, j = t % NDIM;
    size_t lin = (size_t)b * NN2 + t;
    size_t trn = (size_t)b * NN2 + (size_t)j * NDIM + i;
    float wl[4], wt[4];
#pragma unroll
    for (int I = 0; I < 4; ++I) {
      wl[I] = W[(size_t)I * MATSZ + lin];
      wt[I] = W[(size_t)I * MATSZ + trn];
    }
    float d = dDS[lin];
#pragma unroll
    for (int I = 0; I < 4; ++I) {
      aT[I] += wt[I] * d;
#pragma unroll
      for (int J = 0; J < 4; ++J) aO[I][J] += wt[I] * wl[J];
    }
  }
  for (int c = 0; c < 20; ++c) {
    float v = (c < 16) ? aO[c / 4][c % 4] : aT[c - 16];
    red[threadIdx.x] = v; __syncthreads();
    for (int s = 128; s > 0; s >>= 1) { if (threadIdx.x < s) red[threadIdx.x] += red[threadIdx.x + s]; __syncthreads(); }
    if (threadIdx.x == 0) {
      if (c < 16) O[b * 16 + c] = red[0];
      else tr[b * 4 + (c - 16)] = red[0];
    }
    __syncthreads();
  }
}

// coef = inv(O) @ tr via 4x4 Gaussian elimination with partial pivoting
__global__ void k_solve4(const float* __restrict__ O, const float* __restrict__ tr,
                         float* __restrict__ coef) {
  int b = threadIdx.x;
  if (b >= NMOL) return;
  float m[4][5];
  for (int i = 0; i < 4; ++i) {
    for (int j = 0; j < 4; ++j) m[i][j] = O[b * 16 + i * 4 + j];
    m[i][4] = tr[b * 4 + i];
  }
  for (int col = 0; col < 4; ++col) {
    int piv = col;
    float best = fabsf(m[col][col]);
    for (int r = col + 1; r < 4; ++r)
      if (fabsf(m[r][col]) > best) { best = fabsf(m[r][col]); piv = r; }
    if (piv != col)
      for (int c = col; c < 5; ++c) { float t = m[col][c]; m[col][c] = m[piv][c]; m[piv][c] = t; }
    float inv = 1.0f / m[col][col];
    for (int c = col; c < 5; ++c) m[col][c] *= inv;
    for (int r = 0; r < 4; ++r)
      if (r != col) {
        float fct = m[r][col];
        for (int c = col; c < 5; ++c) m[r][c] -= fct * m[col][c];
      }
  }
  for (int i = 0; i < 4; ++i) coef[b * 4 + i] = m[i][4];
}

// dP2dt2, plus per-molecule residual sums for Error
__global__ void __launch_bounds__(256) k_final(float* __restrict__ dP2,
                                               const float* __restrict__ V,
                                               const float* __restrict__ W,
                                               const float* __restrict__ dDS,
                                               const float* __restrict__ coef,
                                               float* __restrict__ num,
                                               float* __restrict__ den) {
  const int b = blockIdx.y;
  const int t = blockIdx.x * blockDim.x + threadIdx.x;
  __shared__ float rn[256], rd[256];
  float na = 0.f, da = 0.f;
  if (t < (int)NN2) {
    size_t idx = (size_t)b * NN2 + t;
    float c0 = coef[b * 4 + 0], c1 = coef[b * 4 + 1];
    float c2 = coef[b * 4 + 2], c3 = coef[b * 4 + 3];
    float dp = -(c0 * V[idx] + c1 * V[MATSZ + idx] + c2 * V[2 * MATSZ + idx] +
                 c3 * V[3 * MATSZ + idx]);
    float id = c0 * W[idx] + c1 * W[MATSZ + idx] + c2 * W[2 * MATSZ + idx] +
               c3 * W[3 * MATSZ + idx];
    float d = dDS[idx];
    dP2[idx] = dp;
    float r = id - d;
    na = r * r;
    da = d * d;
  }
  rn[threadIdx.x] = na; rd[threadIdx.x] = da; __syncthreads();
  for (int s = 128; s > 0; s >>= 1) {
    if (threadIdx.x < s) { rn[threadIdx.x] += rn[threadIdx.x + s]; rd[threadIdx.x] += rd[threadIdx.x + s]; }
    __syncthreads();
  }
  if (threadIdx.x == 0) { atomicAdd(&num[b], rn[0]); atomicAdd(&den[b], rd[0]); }
}

__global__ void k_err(float* __restrict__ err, const float* __restrict__ num,
                      const float* __restrict__ den) {
  int b = threadIdx.x;
  if (b < NMOL) err[b] = sqrtf(num[b] / den[b]);
}

// ---------------------------------------------------------------------------
// Host-side orchestration
// ---------------------------------------------------------------------------
static inline dim3 g1(size_t n, int bs) { return dim3((unsigned)((n + bs - 1) / bs)); }

extern "C" void kernel_launch(void* const* d_in, const int* in_sizes, int n_in,
                              void* d_out, int out_size, void* d_ws, size_t ws_size,
                              hipStream_t stream) {
  (void)n_in; (void)out_size; (void)ws_size;
  const float* P    = (const float*)d_in[0];
  const float* M    = (const float*)d_in[1];
  const float* w    = (const float*)d_in[2];
  const float* gss  = (const float*)d_in[3];
  const float* gpp  = (const float*)d_in[4];
  const float* gsp  = (const float*)d_in[5];
  const float* gp2  = (const float*)d_in[6];
  const float* hsp  = (const float*)d_in[7];
  const int*   maskd = (const int*)d_in[8];
  const int*   mask  = (const int*)d_in[9];
  const int*   idxi  = (const int*)d_in[10];
  const int*   idxj  = (const int*)d_in[11];
  const int*   nocc  = (const int*)d_in[12];
  const int npair = in_sizes[10];

  float* out = (float*)d_out;
  float* Fout   = out;                  // [NMOL, 400, 400]
  float* dP2out = out + MATSZ;          // [NMOL, 400, 400]
  float* Errout = out + 2 * MATSZ;      // [NMOL]
  float* Eout   = out + 2 * MATSZ + NMOL;  // [NMOL]

  // workspace layout (floats)
  float* fws = (float*)d_ws;
  float* Pb   = fws;                    fws += BLKSZ;
  float* Fb   = fws;                    fws += BLKSZ;
  float* Amat = fws;                    fws += MATSZ;
  float* Qmat = fws;                    fws += MATSZ;
  float* dDS  = fws;                    fws += MATSZ;
  float* T1   = fws;                    fws += MATSZ;
  float* T2   = fws;                    fws += MATSZ;
  float* FO1  = fws;                    fws += MATSZ;
  float* Vsl  = fws;                    fws += RANK * MATSZ;
  float* Wsl  = fws;                    fws += RANK * MATSZ;
  float* evec = fws;                    fws += NMOL * NDIM;
  float* focc = fws;                    fws += NMOL * NDIM;
  float* muv  = fws;                    fws += NMOL;
  float* dots = fws;                    fws += NMOL;
  float* n2   = fws;                    fws += NMOL;
  float* Omat = fws;                    fws += NMOL * 16;
  float* trv  = fws;                    fws += NMOL * 4;
  float* coef = fws;                    fws += NMOL * 4;
  float* numb = fws;                    fws += NMOL;
  float* denb = fws;                    fws += NMOL;

  const dim3 gMAT = g1(MATSZ, 256);
  const dim3 gBLK = g1(BLKSZ, 256);
  const dim3 gGEMM((NDIM / 16) * (NDIM / 16), NMOL);  // 625 x 64 tiles

  // ---------------- Fock build: F = fock(M, P) ----------------
  k_blockify<<<gBLK, 256, 0, stream>>>(Pb, P);
  (void)hipMemcpyAsync(Fb, M, BLKSZ * sizeof(float), hipMemcpyDeviceToDevice, stream);
  k_diag_fock<<<g1(NATOM, 256), 256, 0, stream>>>(Fb, Pb, gss, gpp, gsp, gp2, hsp, maskd);
  k_pair_coulomb<<<g1(npair, 256), 256, 0, stream>>>(Fb, Pb, w, maskd, idxi, idxj, npair);
  k_exchange<<<g1(npair, 256), 256, 0, stream>>>(Fb, Pb, w, mask, npair);
  k_unblock_sym<<<gMAT, 256, 0, stream>>>(Fout, Fb);

  // ---------------- eigh(F) via batched parallel Jacobi ----------------
  k_init_eig<<<gMAT, 256, 0, stream>>>(Amat, Qmat, Fout);
  k_jacobi<<<NMOL, 1024, 0, stream>>>(Amat, Qmat);
  k_diag_extract<<<g1(NMOL * NDIM, 256), 256, 0, stream>>>(evec, Amat);

  // ---------------- Fermi occupations, entropy ----------------
  k_fermi<<<NMOL, 512, 0, stream>>>(evec, nocc, focc, Eout, muv);

  // D = 2 Q diag(f) Q^T ; dDS = D - P
  k_colscale<<<gMAT, 256, 0, stream>>>(T1, Qmat, focc);
  k_gemm<0, 1><<<gGEMM, 32, 0, stream>>>(T1, Qmat, T2, 2.0f);
  k_sub<<<gMAT, 256, 0, stream>>>(dDS, T2, P, MATSZ);

  // ---------------- rank-4 Krylov response loop ----------------
  for (int k = 0; k < RANK; ++k) {
    float* Vk = Vsl + (size_t)k * MATSZ;
    const float* dW = (k == 0) ? dDS : (Wsl + (size_t)(k - 1) * MATSZ);
    (void)hipMemcpyAsync(Vk, dW, MATSZ * sizeof(float), hipMemcpyDeviceToDevice, stream);
    for (int j = 0; j < k; ++j) {
      float* Vj = Vsl + (size_t)j * MATSZ;
      k_dot_tr<<<NMOL, 256, 0, stream>>>(Vk, Vj, dots);
      k_proj<<<gMAT, 256, 0, stream>>>(Vk, Vj, dots);
    }
    k_fro<<<NMOL, 256, 0, stream>>>(Vk, n2);
    k_scalenorm<<<gMAT, 256, 0, stream>>>(Vk, n2);

    // FO1 = fock(0, v): two-electron response only
    k_blockify<<<gBLK, 256, 0, stream>>>(Pb, Vk);
    k_fill<<<gBLK, 256, 0, stream>>>(Fb, 0.0f, BLKSZ);
    k_diag_fock<<<g1(NATOM, 256), 256, 0, stream>>>(Fb, Pb, gss, gpp, gsp, gp2, hsp, maskd);
    k_pair_coulomb<<<g1(npair, 256), 256, 0, stream>>>(Fb, Pb, w, maskd, idxi, idxj, npair);
    k_exchange<<<g1(npair, 256), 256, 0, stream>>>(Fb, Pb, w, mask, npair);
    k_unblock_sym<<<gMAT, 256, 0, stream>>>(FO1, Fb);

    // canon_dm_prt: PO1 = 2 Q [ (Q^T FO1 Q) * R ] Q^T
    k_gemm<1, 0><<<gGEMM, 32, 0, stream>>>(Qmat, FO1, T1, 1.0f);  // T1 = Q^T FO1
    k_gemm<0, 0><<<gGEMM, 32, 0, stream>>>(T1, Qmat, T2, 1.0f);   // T2 = F1
    k_rapply<<<gMAT, 256, 0, stream>>>(T2, evec, focc);           // T2 = P1
    k_gemm<0, 0><<<gGEMM, 32, 0, stream>>>(Qmat, T2, T1, 1.0f);   // T1 = Q P1
    k_gemm<0, 1><<<gGEMM, 32, 0, stream>>>(T1, Qmat, T2, 2.0f);   // T2 = PO1

    k_sub<<<gMAT, 256, 0, stream>>>(Wsl + (size_t)k * MATSZ, T2, Vk, MATSZ);  // wk = PO1 - v
  }

  // ---------------- assemble coefficients, outputs ----------------
  k_fill<<<g1(NMOL, 64), 64, 0, stream>>>(numb, 0.0f, NMOL);
  k_fill<<<g1(NMOL, 64), 64, 0, stream>>>(denb, 0.0f, NMOL);
  k_OW<<<NMOL, 256, 0, stream>>>(Wsl, dDS, Omat, trv);
  k_solve4<<<1, NMOL, 0, stream>>>(Omat, trv, coef);
  k_final<<<dim3((unsigned)((NN2 + 255) / 256), NMOL), 256, 0, stream>>>(
      dP2out, Vsl, Wsl, dDS, coef, numb, denb);
  k_err<<<1, NMOL, 0, stream>>>(Errout, numb, denb);
}